// MarkovChainRNN_Model_62912680952080
// MI455X (gfx1250) — compile-verified
//
#include <hip/hip_runtime.h>
#include <cstdint>

// Problem geometry (from the reference)
#define B_TOT   16384
#define T_LEN   2048
#define F_DIM   2
#define POP_INV (1.0f / 1000000.0f)

// Kernel geometry
#define BLOCK   128                       // threads per block = batches per block
#define TILE_T  16                        // timesteps staged per tile
#define NTILES  (T_LEN / TILE_T)          // 128
#define ROW_B   (TILE_T * 8 + 8)          // 136 B: 16 float2 + 8B pad (bank skew: 34 dwords)
#define BUF_B   (BLOCK * ROW_B)           // 17408 B per buffer
#define CHUNKS  16                        // b64 chunks issued per thread per tile

// ---- CDNA5 async global -> LDS copy (ASYNCcnt-tracked) ---------------------
static __device__ __forceinline__ void async_b64_to_lds(uint32_t lds_off,
                                                        uint32_t g_off,
                                                        const void* g_base) {
  // GVS mode: 64-bit SGPR base + 32-bit VGPR byte offset; VDST holds LDS byte addr
  asm volatile("global_load_async_to_lds_b64 %0, %1, %2 offset:0"
               :
               : "v"(lds_off), "v"(g_off), "s"(g_base)
               : "memory");
}

static __device__ __forceinline__ void wait_async0() {
#if __has_builtin(__builtin_amdgcn_s_wait_asynccnt)
  __builtin_amdgcn_s_wait_asynccnt(0);
#else
  asm volatile("s_wait_asynccnt 0" ::: "memory");
#endif
}

// ---------------------------------------------------------------------------
__global__ void __launch_bounds__(BLOCK)
sir_scan_kernel(const float* __restrict__ inputs,      // (B, T, 2)
                const float* __restrict__ init_state,  // (B, 3)
                float* __restrict__ out) {             // 2 x (B, 3) flat
  extern __shared__ __align__(16) unsigned char smem[];

  const int tid   = threadIdx.x;
  const int bBase = blockIdx.x * BLOCK;
  const int b     = bBase + tid;

  // Low 32 bits of a generic pointer to LDS == DS byte address (flat->LDS mapping)
  const uint32_t lds0 = (uint32_t)(uintptr_t)(void*)smem;
  const char* gin = (const char*)inputs;

  // Cooperatively stage one tile: BLOCK rows x (TILE_T steps * 8B), coalesced.
  auto issue = [&](int tile, int bufsel) {
    const uint32_t t0_bytes = (uint32_t)tile * (TILE_T * F_DIM * 4);
    const uint32_t bufbase  = lds0 + (uint32_t)bufsel * BUF_B;
#pragma unroll
    for (int i = 0; i < CHUNKS; ++i) {
      uint32_t f    = (uint32_t)(i * BLOCK + tid);  // flat float2-chunk index
      uint32_t r    = f >> 4;                       // 16 chunks per row
      uint32_t c    = f & 15u;
      uint32_t goff = (uint32_t)(bBase + r) * (uint32_t)(T_LEN * F_DIM * 4)
                      + t0_bytes + c * 8u;
      uint32_t loff = bufbase + r * (uint32_t)ROW_B + c * 8u;
      async_b64_to_lds(loff, goff, gin);
    }
  };

  float S = init_state[b * 3 + 0];
  float I = init_state[b * 3 + 1];
  float R = init_state[b * 3 + 2];

  // Prologue: stage tile 0
  issue(0, 0);
  wait_async0();
  __syncthreads();

  for (int tile = 0; tile < NTILES; ++tile) {
    // Prefetch next tile into the other buffer while computing this one
    if (tile + 1 < NTILES) issue(tile + 1, (tile + 1) & 1);

    const float2* row = reinterpret_cast<const float2*>(
        smem + (size_t)((tile & 1) * BUF_B) + (size_t)tid * ROW_B);
#pragma unroll
    for (int j = 0; j < TILE_T; ++j) {
      float2 u = row[j];                    // (beta, gamma) — conflict-free ds_load_b64
      float new_inf = u.x * S * I * POP_INV;
      float new_rec = u.y * I;
      S = S - new_inf;
      I = I + new_inf - new_rec;
      R = R + new_rec;
    }

    wait_async0();      // my async loads for next tile landed in LDS
    __syncthreads();    // everyone's loads landed; everyone done reading current buf
  }

  // Tuple output: final_state twice, concatenated flat
  out[b * 3 + 0] = S;
  out[b * 3 + 1] = I;
  out[b * 3 + 2] = R;
  out[(size_t)B_TOT * 3 + b * 3 + 0] = S;
  out[(size_t)B_TOT * 3 + b * 3 + 1] = I;
  out[(size_t)B_TOT * 3 + b * 3 + 2] = R;
}

// ---------------------------------------------------------------------------
extern "C" void kernel_launch(void* const* d_in, const int* in_sizes, int n_in,
                              void* d_out, int out_size, void* d_ws, size_t ws_size,
                              hipStream_t stream) {
  const float* inputs = (const float*)d_in[0];      // (B, T, 2) f32
  const float* init   = (const float*)d_in[1];      // (B, 3)    f32
  float* out          = (float*)d_out;              // 2*B*3     f32

  dim3 grid(B_TOT / BLOCK);   // 128 blocks
  dim3 block(BLOCK);          // 128 threads (4 waves, wave32)
  size_t smem = 2 * BUF_B;    // 34,816 B double-buffered staging
  sir_scan_kernel<<<grid, block, smem, stream>>>(inputs, init, out);
}